// TwoBranchGNN_Normal_Neg_A_34437047780020
// MI455X (gfx1250) — compile-verified
//
#include <hip/hip_runtime.h>

// ---------------- problem constants (match reference) ----------------
#define NN      50000
#define EE      800000
#define IN_DIM  256
#define H1      256
#define H2P     64      // H2=62 padded to 64
#define OUTD    64

typedef float v2f __attribute__((ext_vector_type(2)));
typedef float v8f __attribute__((ext_vector_type(8)));

// ---------------- small elementwise kernels ----------------
__global__ void init_const_kernel(float* p, float v, int n) {
    int t = blockIdx.x * blockDim.x + threadIdx.x;
    if (t < n) p[t] = v;
}

__global__ void deg_atomic_kernel(const int* __restrict__ dstIdx, float* __restrict__ deg, int e) {
    int t = blockIdx.x * blockDim.x + threadIdx.x;
    if (t < e) atomicAdd(&deg[dstIdx[t]], 1.0f);
}

__global__ void rsqrt_ip_kernel(float* p, int n) {
    int t = blockIdx.x * blockDim.x + threadIdx.x;
    if (t < n) p[t] = rsqrtf(p[t]);
}

// zero-pad copy: src [sr x sc] -> dst [dr x dc] (row-major), used for biases
__global__ void pad_kernel(const float* __restrict__ src, float* __restrict__ dst,
                           int sr, int sc, int dr, int dc) {
    int t = blockIdx.x * blockDim.x + threadIdx.x;
    if (t >= dr * dc) return;
    int r = t / dc, c = t % dc;
    dst[t] = (r < sr && c < sc) ? src[r * sc + c] : 0.0f;
}

// transpose + zero-pad: src [sr x sc] row-major -> dst [dc x Kp]  (dst[c*Kp + r] = src[r][c])
// Makes each B fragment (two K-consecutive elements of one output column) a single b64 load.
__global__ void transpose_pad_kernel(const float* __restrict__ src, float* __restrict__ dst,
                                     int sr, int sc, int Kp, int dc) {
    int t = blockIdx.x * blockDim.x + threadIdx.x;
    if (t >= dc * Kp) return;
    int c = t / Kp, r = t % Kp;
    dst[t] = (r < sr && c < sc) ? src[r * sc + c] : 0.0f;
}

// ---------------- fp32 WMMA GEMM: C[M x Dout] = A[M x K] @ W[K x Dout] (+bias) ----------------
// Weights passed TRANSPOSED: Wt[Dout x K], Wt[c*K + k] = W[k][c].
// One wave computes a 32x32 C tile (2x2 subtiles, 4 accumulators) -> per k-step:
// 2 A-loads + 2 B-loads (all b64) feed 4 V_WMMA_F32_16X16X4_F32 (1 load per WMMA).
// The k-loop uses a wave-uniform counter (lane-dependent `half` is folded into the
// fragment base pointers) so the loop is a scalar branch and EXEC stays all-1s at
// every WMMA, as the ISA requires.
// Lane layout (ISA 7.12.2):
//   A 16x4 : lanes 0-15 hold M=lane, K=k+{0,1}; lanes 16-31 hold M=lane-16, K=k+{2,3}
//   B 4x16 : lanes 0-15 hold N=lane rows K=k+{0,1}; lanes 16-31 N=lane-16 rows K=k+{2,3}
//   C 16x16: VGPR j -> M = j + 8*(lane>=16), N = lane&15
__global__ void gemm_wmma_f32_kernel(const float* __restrict__ A, const float* __restrict__ Wt,
                                     const float* __restrict__ bias, float* __restrict__ C,
                                     int M, int K, int Dout) {
    const int lane = threadIdx.x & 31;
    const int wv   = threadIdx.x >> 5;
    const int row0 = (blockIdx.x * 4 + wv) * 32;   // wave-uniform
    if (row0 >= M) return;                          // whole-wave exit
    const int col0 = blockIdx.y * 32;

    const int l15  = lane & 15;
    const int half = lane >> 4;                     // selects K sub-pair {0,1} vs {2,3}

    const int rA0 = row0 + l15;
    int rA1 = row0 + 16 + l15;
    if (rA1 >= M) rA1 = M - 1;                      // clamp loads; stores guarded below

    // fold lane-dependent `half` into the base pointers -> uniform loop counter
    const v2f* A0p = (const v2f*)(A  + (size_t)rA0 * K) + half;
    const v2f* A1p = (const v2f*)(A  + (size_t)rA1 * K) + half;
    const v2f* B0p = (const v2f*)(Wt + (size_t)(col0 + l15) * K) + half;
    const v2f* B1p = (const v2f*)(Wt + (size_t)(col0 + 16 + l15) * K) + half;

    v8f acc00 = {}, acc01 = {}, acc10 = {}, acc11 = {};
    const int steps = K >> 2;                       // wave-uniform trip count
    for (int i = 0; i < steps; ++i) {               // k = 4*i; fragment index = k/2 + half
        v2f a0 = A0p[2 * i];
        v2f a1 = A1p[2 * i];
        v2f b0 = B0p[2 * i];
        v2f b1 = B1p[2 * i];
        acc00 = __builtin_amdgcn_wmma_f32_16x16x4_f32(false, a0, false, b0, (short)0, acc00, false, false);
        acc01 = __builtin_amdgcn_wmma_f32_16x16x4_f32(false, a0, false, b1, (short)0, acc01, false, false);
        acc10 = __builtin_amdgcn_wmma_f32_16x16x4_f32(false, a1, false, b0, (short)0, acc10, false, false);
        acc11 = __builtin_amdgcn_wmma_f32_16x16x4_f32(false, a1, false, b1, (short)0, acc11, false, false);
    }

    const float bv0 = bias ? bias[col0 + l15] : 0.0f;
    const float bv1 = bias ? bias[col0 + 16 + l15] : 0.0f;

    {   // M subtile 0 (always fully valid: M % 16 == 0 and row0 < M)
        float* Cp = C + (size_t)(row0 + half * 8) * Dout + col0;
#pragma unroll
        for (int j = 0; j < 8; ++j) {
            Cp[(size_t)j * Dout + l15]      = acc00[j] + bv0;
            Cp[(size_t)j * Dout + 16 + l15] = acc01[j] + bv1;
        }
    }
    if (row0 + 16 < M) {   // M subtile 1
        float* Cp = C + (size_t)(row0 + 16 + half * 8) * Dout + col0;
#pragma unroll
        for (int j = 0; j < 8; ++j) {
            Cp[(size_t)j * Dout + l15]      = acc10[j] + bv0;
            Cp[(size_t)j * Dout + 16 + l15] = acc11[j] + bv1;
        }
    }
}

// ---------------- edge scatter-add: agg[dst] += dinv[src]*dinv[dst] * tmp[src] ----------------
__global__ void scatter_kernel(const float* __restrict__ tmp, const int* __restrict__ srcIdx,
                               const int* __restrict__ dstIdx, const float* __restrict__ dinv,
                               float* __restrict__ agg, int e_cnt, int n_cnt, int D) {
    const int dq = D >> 2;
    long long t = (long long)blockIdx.x * blockDim.x + threadIdx.x;
    long long total = (long long)(e_cnt + n_cnt) * dq;
    if (t >= total) return;
    int e = (int)(t / dq);
    int f = (int)(t % dq) << 2;
    int s, d;
    if (e < e_cnt) { s = srcIdx[e]; d = dstIdx[e]; }
    else           { s = d = e - e_cnt; }
    const float w = dinv[s] * dinv[d];
    const float4 v = *(const float4*)(tmp + (size_t)s * D + f);
    float* ap = agg + (size_t)d * D + f;
    atomicAdd(ap + 0, w * v.x);
    atomicAdd(ap + 1, w * v.y);
    atomicAdd(ap + 2, w * v.z);
    atomicAdd(ap + 3, w * v.w);
}

// ---------------- combine: out = base + scale * act(agg + bias) ----------------
__global__ void combine_kernel(const float* __restrict__ base, const float* __restrict__ agg,
                               const float* __restrict__ bias, float* __restrict__ out,
                               float scale, int do_relu, long long total4, int D) {
    long long t = (long long)blockIdx.x * blockDim.x + threadIdx.x;
    if (t >= total4) return;
    long long off = t << 2;
    int f = (int)(off % D);
    float4 a = *(const float4*)(agg + off);
    float4 b = *(const float4*)(base + off);
    float v0 = a.x + bias[f + 0];
    float v1 = a.y + bias[f + 1];
    float v2 = a.z + bias[f + 2];
    float v3 = a.w + bias[f + 3];
    if (do_relu) {
        v0 = fmaxf(v0, 0.0f); v1 = fmaxf(v1, 0.0f);
        v2 = fmaxf(v2, 0.0f); v3 = fmaxf(v3, 0.0f);
    }
    float4 r;
    r.x = b.x + scale * v0; r.y = b.y + scale * v1;
    r.z = b.z + scale * v2; r.w = b.w + scale * v3;
    *(float4*)(out + off) = r;
}

// mode 0: out = 0.5*x ; mode 1: out -= 0.5*x
__global__ void final_kernel(const float* __restrict__ x, float* __restrict__ out,
                             int mode, long long n) {
    long long t = (long long)blockIdx.x * blockDim.x + threadIdx.x;
    if (t >= n) return;
    if (mode == 0) out[t] = 0.5f * x[t];
    else           out[t] -= 0.5f * x[t];
}

// ---------------- host side ----------------
struct BranchP {
    const float *l1W, *l1b, *l2W, *l2b, *l3W, *l3b;
    const float *gW[6], *gb[6];
};

// prepared (transposed / padded) parameters in workspace
struct PrepP {
    float *l1Wt, *l2Wt, *l3Wt;   // [Dout x K] transposed, zero-padded
    float *gWt[6];
    float *l2bp, *g2bp;          // padded biases (62 -> 64)
};

static void fill_insertion(void* const* d, int base, BranchP& P) {
    P.l1W = (const float*)d[base + 0];  P.l1b = (const float*)d[base + 1];
    P.l2W = (const float*)d[base + 2];  P.l2b = (const float*)d[base + 3];
    P.l3W = (const float*)d[base + 4];  P.l3b = (const float*)d[base + 5];
    for (int i = 0; i < 6; ++i) {
        P.gW[i] = (const float*)d[base + 6 + 2 * i];
        P.gb[i] = (const float*)d[base + 7 + 2 * i];
    }
}

static void fill_sorted(void* const* d, int base, BranchP& P) {
    for (int i = 0; i < 6; ++i) {
        P.gW[i] = (const float*)d[base + 2 * i];
        P.gb[i] = (const float*)d[base + 1 + 2 * i];
    }
    P.l1W = (const float*)d[base + 12]; P.l1b = (const float*)d[base + 13];
    P.l2W = (const float*)d[base + 14]; P.l2b = (const float*)d[base + 15];
    P.l3W = (const float*)d[base + 16]; P.l3b = (const float*)d[base + 17];
}

static inline void launch_gemm(hipStream_t s, const float* A, const float* Wt,
                               const float* bias, float* C, int K, int Dout) {
    dim3 grid((NN + 127) / 128, Dout / 32);
    gemm_wmma_f32_kernel<<<grid, 128, 0, s>>>(A, Wt, bias, C, NN, K, Dout);
}

static inline void launch_combine(hipStream_t s, const float* base, const float* agg,
                                  const float* bias, float* out, float scale, int relu, int D) {
    long long total4 = (long long)NN * D / 4;
    combine_kernel<<<(unsigned)((total4 + 255) / 256), 256, 0, s>>>(base, agg, bias, out,
                                                                    scale, relu, total4, D);
}

static void gcn_agg(hipStream_t s, const float* h, const float* gWt,
                    const int* srcI, const int* dstI, const float* dinv,
                    float* tmp, float* agg, int K, int D) {
    launch_gemm(s, h, gWt, nullptr, tmp, K, D);
    hipMemsetAsync(agg, 0, (size_t)NN * D * sizeof(float), s);
    long long total = (long long)(EE + NN) * (D / 4);
    scatter_kernel<<<(unsigned)((total + 255) / 256), 256, 0, s>>>(tmp, srcI, dstI, dinv,
                                                                   agg, EE, NN, D);
}

static void run_branch(hipStream_t s, const float* x, const int* edges, const float* dinv,
                       const BranchP& P, const PrepP& R,
                       float* B0, float* B1, float* B2, float* B3,
                       float* d_out, int mode) {
    const int* srcI = edges;
    const int* dstI = edges + EE;

    // layer 1 (256 -> 256)
    launch_gemm(s, x, R.l1Wt, P.l1b, B0, IN_DIM, H1);                // x1l -> B0
    gcn_agg(s, B0, R.gWt[0], srcI, dstI, dinv, B1, B2, H1, H1);
    launch_combine(s, B0, B2, P.gb[0], B0, 1.0f, 1, H1);             // x1 -> B0
    // layer 2 (256 -> 62 padded to 64)
    launch_gemm(s, B0, R.l2Wt, R.l2bp, B3, H1, H2P);                 // x2l -> B3
    gcn_agg(s, B3, R.gWt[1], srcI, dstI, dinv, B1, B2, H2P, H2P);
    launch_combine(s, B3, B2, R.g2bp, B3, 1.0f, 1, H2P);             // x2 -> B3
    // layer 3 (62p -> 64)
    launch_gemm(s, B3, R.l3Wt, P.l3b, B0, H2P, OUTD);                // x3l -> B0
    gcn_agg(s, B0, R.gWt[2], srcI, dstI, dinv, B1, B2, OUTD, OUTD);
    launch_combine(s, B0, B2, P.gb[2], B0, 0.5f, 1, OUTD);           // x3 -> B0
    // layer 4
    gcn_agg(s, B0, R.gWt[3], srcI, dstI, dinv, B1, B2, OUTD, OUTD);
    launch_combine(s, B0, B2, P.gb[3], B3, 0.5f, 1, OUTD);           // x4 -> B3
    // layer 5
    gcn_agg(s, B3, R.gWt[4], srcI, dstI, dinv, B1, B2, OUTD, OUTD);
    launch_combine(s, B3, B2, P.gb[4], B0, 0.25f, 1, OUTD);          // x5 -> B0
    // layer 6 (no relu)
    gcn_agg(s, B0, R.gWt[5], srcI, dstI, dinv, B1, B2, OUTD, OUTD);
    launch_combine(s, B0, B2, P.gb[5], B3, 0.25f, 0, OUTD);          // x6 -> B3

    long long n = (long long)NN * OUTD;
    final_kernel<<<(unsigned)((n + 255) / 256), 256, 0, s>>>(B3, d_out, mode, n);
}

extern "C" void kernel_launch(void* const* d_in, const int* in_sizes, int n_in,
                              void* d_out, int out_size, void* d_ws, size_t ws_size,
                              hipStream_t stream) {
    (void)n_in; (void)out_size; (void)ws_size;

    // Input-order detection: dict-insertion (x first) vs jax-sorted (edge_neg first)
    const bool ins = (in_sizes[0] == NN * IN_DIM);
    const float* x;
    const int *ep, *en;
    BranchP Pp, Pn;
    if (ins) {
        x  = (const float*)d_in[0];
        ep = (const int*)d_in[1];
        en = (const int*)d_in[2];
        fill_insertion(d_in, 3, Pp);
        fill_insertion(d_in, 21, Pn);
    } else {
        en = (const int*)d_in[0];
        ep = (const int*)d_in[1];
        fill_sorted(d_in, 2, Pn);
        fill_sorted(d_in, 20, Pp);
        x  = (const float*)d_in[38];
    }

    // workspace layout (floats)
    float* ws = (float*)d_ws;
    const size_t NB = (size_t)NN * 256;
    float* B0 = ws;
    float* B1 = ws + NB;
    float* B2 = ws + 2 * NB;
    float* B3 = ws + 3 * NB;
    float* dinvP = ws + 4 * NB;
    float* dinvN = dinvP + NN;
    float* prepBase = dinvN + NN;
    // per-branch prepared params: l1Wt 65536 | l2Wt 16384 | l3Wt 4096 | g1Wt 65536 |
    // g2..g6Wt 5*4096 | l2bp 64 | g2bp 64  => 172160 floats
    const size_t PER = 172160;
    PrepP R[2];
    for (int b = 0; b < 2; ++b) {
        float* base = prepBase + b * PER;
        R[b].l1Wt   = base;
        R[b].l2Wt   = base + 65536;
        R[b].l3Wt   = base + 81920;
        R[b].gWt[0] = base + 86016;
        R[b].gWt[1] = base + 151552;
        R[b].gWt[2] = base + 155648;
        R[b].gWt[3] = base + 159744;
        R[b].gWt[4] = base + 163840;
        R[b].gWt[5] = base + 167936;
        R[b].l2bp   = base + 172032;
        R[b].g2bp   = base + 172096;
    }

    // prepare weights: transpose (+ zero-pad 62 -> 64) so B fragments are b64 loads
    const BranchP* PB[2] = { &Pp, &Pn };
    for (int b = 0; b < 2; ++b) {
        const BranchP& P = *PB[b];
        transpose_pad_kernel<<<(65536 + 255) / 256, 256, 0, stream>>>(P.l1W, R[b].l1Wt, 256, 256, 256, 256);
        transpose_pad_kernel<<<(16384 + 255) / 256, 256, 0, stream>>>(P.l2W, R[b].l2Wt, 256, 62, 256, 64);
        transpose_pad_kernel<<<(4096  + 255) / 256, 256, 0, stream>>>(P.l3W, R[b].l3Wt, 62, 64, 64, 64);
        transpose_pad_kernel<<<(65536 + 255) / 256, 256, 0, stream>>>(P.gW[0], R[b].gWt[0], 256, 256, 256, 256);
        transpose_pad_kernel<<<(4096  + 255) / 256, 256, 0, stream>>>(P.gW[1], R[b].gWt[1], 62, 62, 64, 64);
        for (int i = 2; i < 6; ++i)
            transpose_pad_kernel<<<(4096 + 255) / 256, 256, 0, stream>>>(P.gW[i], R[b].gWt[i], 64, 64, 64, 64);
        pad_kernel<<<1, 64, 0, stream>>>(P.l2b,  R[b].l2bp, 1, 62, 1, 64);
        pad_kernel<<<1, 64, 0, stream>>>(P.gb[1], R[b].g2bp, 1, 62, 1, 64);
    }

    // degrees -> dinv (self loop contributes 1)
    init_const_kernel<<<(NN + 255) / 256, 256, 0, stream>>>(dinvP, 1.0f, NN);
    deg_atomic_kernel<<<(EE + 255) / 256, 256, 0, stream>>>(ep + EE, dinvP, EE);
    rsqrt_ip_kernel<<<(NN + 255) / 256, 256, 0, stream>>>(dinvP, NN);
    init_const_kernel<<<(NN + 255) / 256, 256, 0, stream>>>(dinvN, 1.0f, NN);
    deg_atomic_kernel<<<(EE + 255) / 256, 256, 0, stream>>>(en + EE, dinvN, EE);
    rsqrt_ip_kernel<<<(NN + 255) / 256, 256, 0, stream>>>(dinvN, NN);

    float* out = (float*)d_out;
    run_branch(stream, x, ep, dinvP, Pp, R[0], B0, B1, B2, B3, out, /*set*/0);
    run_branch(stream, x, en, dinvN, Pn, R[1], B0, B1, B2, B3, out, /*sub*/1);
}